// GNNBranch_34857954574434
// MI455X (gfx1250) — compile-verified
//
#include <hip/hip_runtime.h>

// Problem constants (match reference)
#define N_NODES   50000
#define N_EDGES   800000
#define G_GRAPHS  64
#define F_IN      128
#define HIDF      128
#define H_HEADS   8
#define D_HEAD    16
#define NEG_SLOPE 0.01f
#define LN_EPS    1e-5f
#define SM_EPS    1e-16f
#define MT_TOTAL  (N_NODES / 16)   // 3125 row tiles

typedef __attribute__((ext_vector_type(2))) float    v2f;
typedef __attribute__((ext_vector_type(8))) float    v8f;
typedef __attribute__((ext_vector_type(4))) unsigned v4u;
typedef __attribute__((ext_vector_type(8))) int      v8i;
typedef __attribute__((ext_vector_type(4))) int      v4i;

typedef __attribute__((address_space(3))) const float lds_cf;

// Monotone mapping float -> uint so unsigned max == float max.
__device__ __forceinline__ unsigned f2key(float f) {
  unsigned u = __float_as_uint(f);
  return (u & 0x80000000u) ? ~u : (u | 0x80000000u);
}
__device__ __forceinline__ float key2f(unsigned k) {
  return (k & 0x80000000u) ? __uint_as_float(k & 0x7fffffffu)
                           : __uint_as_float(~k);
}
#define KEY_NEG_INF 0x007fffffu  // f2key(-inf)

// ---------------------------------------------------------------------------
// Y[N,128] = X[N,128] @ W[128,128] + bias[128], fp32 WMMA 16x16x4.
// Block = 8 waves; blockIdx.y picks the 16-wide column tile. The W column
// tile (128x16 f32, 8 KB) is staged into LDS once per block by the Tensor
// Data Mover (tensor_load_to_lds), then all 8 waves feed their WMMA
// B-fragments from LDS (ds_load) while streaming A-fragments from global.
// The TDM writes LDS behind the compiler's back, so the tile pointer is
// laundered (as an address-space-3 pointer, to keep ds_load addressing)
// through an asm block with a memory clobber.
// ---------------------------------------------------------------------------
__global__ __launch_bounds__(256)
void gemm_x128_wmma(const float* __restrict__ X, const float* __restrict__ W,
                    const float* __restrict__ B, float* __restrict__ Y) {
  __shared__ float wtile[F_IN * 16];  // [k][c] row-major, 8 KB

  const int lane = threadIdx.x & 31;
  const int wave = threadIdx.x >> 5;
  const int nt   = blockIdx.y;
  const int col0 = nt * 16;

  if (wave == 0) {
    // TDM 2-D descriptor: tile 16 (contiguous, dim0) x 128 rows (dim1),
    // row stride 128 floats, 4-byte elements, into wtile's LDS offset.
    unsigned lds_base = (unsigned)(uintptr_t)(void*)wtile;  // LDS byte offset
    unsigned long long gaddr = (unsigned long long)(const void*)(W + col0);
    v4u g0;
    g0.x = 1u;                                   // count=1, user descriptor
    g0.y = lds_base;                             // lds_addr
    g0.z = (unsigned)(gaddr & 0xffffffffu);      // global_addr[31:0]
    g0.w = (unsigned)((gaddr >> 32) & 0x01ffffffu) | (2u << 30);  // type=2
    v8i g1;
    g1[0] = (int)(2u << 16);       // data_size = 2 -> 4 bytes
    g1[1] = (int)(128u << 16);     // tensor_dim0 = 128 (bits 79:48 low half)
    g1[2] = (int)(128u << 16);     // tensor_dim1 = 128 (bits 111:80 low half)
    g1[3] = (int)(16u << 16);      // tile_dim0 = 16   (bits 127:112)
    g1[4] = 128;                   // tile_dim1 = 128  (bits 143:128)
    g1[5] = 128;                   // tensor_dim0_stride = 128 (bits 191:160)
    g1[6] = 0;
    g1[7] = 0;
    v4i gz4 = {0, 0, 0, 0};                      // groups 2/3 unused (2-D)
    v8i gz8 = {0, 0, 0, 0, 0, 0, 0, 0};
    __builtin_amdgcn_tensor_load_to_lds(g0, g1, gz4, gz4, gz8, 0);
    __builtin_amdgcn_s_wait_tensorcnt(0);
  }
  __syncthreads();                 // LDS tile visible to all 8 waves

  // Launder the tile pointer in LDS address space: the asm may have written
  // LDS (memory clobber), so reads through wt are defined, stay alive, and
  // still lower to ds_load (32-bit LDS addressing).
  lds_cf* wt = (lds_cf*)wtile;
  asm volatile("" : "+v"(wt) : : "memory");

  const int  mt     = blockIdx.x * 8 + wave;     // row tile
  const bool active = (mt < MT_TOTAL);           // wave-uniform
  const int  mtc    = active ? mt : (MT_TOTAL - 1);
  const int  half   = lane >> 4;                 // 0: lanes 0-15, 1: 16-31
  const int  mr     = lane & 15;
  const int  col    = col0 + mr;

  const float bb = B[col];
  v8f acc;
#pragma unroll
  for (int j = 0; j < 8; ++j) acc[j] = bb;       // bias depends on N only

  const float* xrow = X + (size_t)(mtc * 16 + mr) * F_IN;
#pragma unroll 4
  for (int k0 = 0; k0 < F_IN; k0 += 4) {
    // A 16x4: lanes 0-15 hold K=k0,k0+1; lanes 16-31 hold K=k0+2,k0+3
    v2f a, b;
    a.x = xrow[k0 + 2 * half];
    a.y = xrow[k0 + 2 * half + 1];
    // B 4x16 from LDS tile (row K striped across lanes)
    b.x = wt[(k0 + 2 * half) * 16 + mr];
    b.y = wt[(k0 + 2 * half + 1) * 16 + mr];
    acc = __builtin_amdgcn_wmma_f32_16x16x4_f32(false, a, false, b,
                                                (short)0, acc, false, false);
  }
  // C/D: VGPR j -> row j (lanes 0-15) / row j+8 (lanes 16-31), col = lane&15
  if (active) {
#pragma unroll
    for (int j = 0; j < 8; ++j)
      Y[(size_t)(mtc * 16 + half * 8 + j) * HIDF + col] = acc[j];
  }
}

// ---------------------------------------------------------------------------
__global__ void init_nh(unsigned* __restrict__ mkey, float* __restrict__ ssum) {
  int i = blockIdx.x * blockDim.x + threadIdx.x;
  if (i >= N_NODES * H_HEADS) return;
  mkey[i] = KEY_NEG_INF;
  ssum[i] = 0.f;
}

__global__ void zero_f(float* __restrict__ p, int n) {
  int i = blockIdx.x * blockDim.x + threadIdx.x;
  if (i < n) p[i] = 0.f;
}

// Pass 1: scores[e,h] = <q[dst,h,:], k[src,h,:]> / sqrt(D); segment max by dst.
__global__ void edge_scores(const float* __restrict__ q, const float* __restrict__ k,
                            const int* __restrict__ src, const int* __restrict__ dst,
                            float* __restrict__ sc, unsigned* __restrict__ mkey) {
  int idx = blockIdx.x * blockDim.x + threadIdx.x;
  if (idx >= N_EDGES * H_HEADS) return;
  int e = idx >> 3, h = idx & 7;
  int s = src[e], d = dst[e];
  const float4* qp = (const float4*)(q + (size_t)d * HIDF + h * D_HEAD);
  const float4* kp = (const float4*)(k + (size_t)s * HIDF + h * D_HEAD);
  float acc = 0.f;
#pragma unroll
  for (int i = 0; i < 4; ++i) {
    float4 a = qp[i], b = kp[i];
    acc += a.x * b.x + a.y * b.y + a.z * b.z + a.w * b.w;
  }
  acc *= 0.25f;  // 1/sqrt(16)
  sc[idx] = acc;
  atomicMax(&mkey[d * H_HEADS + h], f2key(acc));
}

// Pass 2: e = exp(score - m[dst]); segment sum by dst.
__global__ void edge_exp(const int* __restrict__ dst, float* __restrict__ sc,
                         const unsigned* __restrict__ mkey, float* __restrict__ ssum) {
  int idx = blockIdx.x * blockDim.x + threadIdx.x;
  if (idx >= N_EDGES * H_HEADS) return;
  int e = idx >> 3, h = idx & 7;
  int d = dst[e];
  float m  = key2f(mkey[d * H_HEADS + h]);
  float ev = expf(sc[idx] - m);
  sc[idx] = ev;
  atomicAdd(&ssum[d * H_HEADS + h], ev);
}

// Pass 3: alpha = e/(s[dst]+eps); out[dst] += alpha * v[src]  (into skip buffer).
__global__ void edge_scatter(const int* __restrict__ src, const int* __restrict__ dst,
                             const float* __restrict__ sc, const float* __restrict__ ssum,
                             const float* __restrict__ v, float* __restrict__ alpha_out,
                             float* __restrict__ hout) {
  int idx = blockIdx.x * blockDim.x + threadIdx.x;
  if (idx >= N_EDGES * H_HEADS) return;
  int e = idx >> 3, h = idx & 7;
  int s = src[e], d = dst[e];
  float alpha = sc[idx] / (ssum[d * H_HEADS + h] + SM_EPS);
  alpha_out[idx] = alpha;
  const float* vp = v + (size_t)s * HIDF + h * D_HEAD;
  float* op = hout + (size_t)d * HIDF + h * D_HEAD;
#pragma unroll
  for (int i = 0; i < D_HEAD; ++i) atomicAdd(&op[i], alpha * vp[i]);
}

// Leaky-ReLU in place + per-graph sum / sumsq / count.
__global__ void relu_stats(float* __restrict__ h, const int* __restrict__ batch,
                           float* __restrict__ gsum, float* __restrict__ gsq,
                           float* __restrict__ gcnt) {
  int n = blockIdx.x * blockDim.x + threadIdx.x;
  if (n >= N_NODES) return;
  int g = batch[n];
  float* row = h + (size_t)n * HIDF;
  float s = 0.f, s2 = 0.f;
#pragma unroll 4
  for (int f = 0; f < HIDF; ++f) {
    float val = row[f];
    val = (val >= 0.f) ? val : NEG_SLOPE * val;
    row[f] = val;
    s += val;
    s2 += val * val;
  }
  atomicAdd(&gsum[g], s);
  atomicAdd(&gsq[g], s2);
  atomicAdd(&gcnt[g], 1.0f);
}

__global__ void finalize_stats(const float* __restrict__ gsum, const float* __restrict__ gsq,
                               const float* __restrict__ gcnt, float* __restrict__ gmean,
                               float* __restrict__ ginv) {
  int g = threadIdx.x;
  if (g >= G_GRAPHS) return;
  float c = gcnt[g] * (float)HIDF;
  if (c > 0.f) {
    float mean = gsum[g] / c;
    float var  = gsq[g] / c - mean * mean;
    gmean[g] = mean;
    ginv[g]  = rsqrtf(var + LN_EPS);
  } else {
    gmean[g] = 0.f;
    ginv[g]  = 0.f;
  }
}

__global__ void apply_ln(float* __restrict__ h, const int* __restrict__ batch,
                         const float* __restrict__ gmean, const float* __restrict__ ginv,
                         const float* __restrict__ gamma, const float* __restrict__ beta) {
  int idx = blockIdx.x * blockDim.x + threadIdx.x;
  if (idx >= N_NODES * HIDF) return;
  int n = idx >> 7, f = idx & 127;
  int g = batch[n];
  h[idx] = (h[idx] - gmean[g]) * ginv[g] * gamma[f] + beta[f];
}

__global__ void pool_acc(const float* __restrict__ h, const int* __restrict__ batch,
                         float* __restrict__ feat) {
  int idx = blockIdx.x * blockDim.x + threadIdx.x;
  if (idx >= N_NODES * HIDF) return;
  int n = idx >> 7, f = idx & 127;
  atomicAdd(&feat[batch[n] * HIDF + f], h[idx]);
}

__global__ void pool_div(float* __restrict__ feat, const float* __restrict__ gcnt) {
  int idx = blockIdx.x * blockDim.x + threadIdx.x;
  if (idx >= G_GRAPHS * HIDF) return;
  int g = idx >> 7;
  feat[idx] /= fmaxf(gcnt[g], 1.0f);
}

// ---------------------------------------------------------------------------
extern "C" void kernel_launch(void* const* d_in, const int* in_sizes, int n_in,
                              void* d_out, int out_size, void* d_ws, size_t ws_size,
                              hipStream_t stream) {
  const float* x     = (const float*)d_in[0];
  const int*   ei    = (const int*)d_in[1];
  const int*   src   = ei;
  const int*   dst   = ei + N_EDGES;
  const int*   batch = (const int*)d_in[2];
  const float *Wq1 = (const float*)d_in[3],  *bq1 = (const float*)d_in[4];
  const float *Wk1 = (const float*)d_in[5],  *bk1 = (const float*)d_in[6];
  const float *Wv1 = (const float*)d_in[7],  *bv1 = (const float*)d_in[8];
  const float *Ws1 = (const float*)d_in[9],  *bs1 = (const float*)d_in[10];
  const float *g1  = (const float*)d_in[11], *be1 = (const float*)d_in[12];
  const float *Wq2 = (const float*)d_in[13], *bq2 = (const float*)d_in[14];
  const float *Wk2 = (const float*)d_in[15], *bk2 = (const float*)d_in[16];
  const float *Wv2 = (const float*)d_in[17], *bv2 = (const float*)d_in[18];
  const float *Ws2 = (const float*)d_in[19], *bs2 = (const float*)d_in[20];
  const float *g2  = (const float*)d_in[21], *be2 = (const float*)d_in[22];

  const size_t NH = (size_t)N_NODES * HIDF;    // 6.4M
  const size_t EH = (size_t)N_EDGES * H_HEADS; // 6.4M
  float* ws = (float*)d_ws;
  float*    q    = ws;
  float*    k    = q + NH;
  float*    v    = k + NH;
  float*    h1   = v + NH;      // skip1 -> layer-1 node features (in place)
  float*    h2   = h1 + NH;     // skip2 -> layer-2 node features (in place)
  float*    sc   = h2 + NH;     // scores / exp values (reused per layer)
  unsigned* mkey = (unsigned*)(sc + EH);
  float*    ssum = (float*)(mkey + (size_t)N_NODES * H_HEADS);
  float*    gsum = ssum + (size_t)N_NODES * H_HEADS;
  float*    gsq  = gsum + G_GRAPHS;
  float*    gcnt = gsq + G_GRAPHS;
  float*    gmean= gcnt + G_GRAPHS;
  float*    ginv = gmean + G_GRAPHS;

  float* feat   = (float*)d_out;              // [64,128]
  float* alpha1 = feat + G_GRAPHS * HIDF;     // [E,8]
  float* alpha2 = alpha1 + EH;                // [E,8]

  const dim3 gemmGrid((MT_TOTAL + 7) / 8, HIDF / 16);
  const int eBlk  = (int)((EH + 255) / 256);
  const int nhBlk = (N_NODES * H_HEADS + 255) / 256;
  const int nfBlk = (int)((NH + 255) / 256);
  const int nBlk  = (N_NODES + 255) / 256;

  auto conv = [&](const float* xin,
                  const float* Wq, const float* bq, const float* Wk_, const float* bk_,
                  const float* Wv_, const float* bv_, const float* Ws_, const float* bs_,
                  const float* gam, const float* bet, float* hout, float* aout) {
    gemm_x128_wmma<<<gemmGrid, 256, 0, stream>>>(xin, Wq, bq, q);
    gemm_x128_wmma<<<gemmGrid, 256, 0, stream>>>(xin, Wk_, bk_, k);
    gemm_x128_wmma<<<gemmGrid, 256, 0, stream>>>(xin, Wv_, bv_, v);
    gemm_x128_wmma<<<gemmGrid, 256, 0, stream>>>(xin, Ws_, bs_, hout);  // skip
    init_nh<<<nhBlk, 256, 0, stream>>>(mkey, ssum);
    edge_scores<<<eBlk, 256, 0, stream>>>(q, k, src, dst, sc, mkey);
    edge_exp<<<eBlk, 256, 0, stream>>>(dst, sc, mkey, ssum);
    edge_scatter<<<eBlk, 256, 0, stream>>>(src, dst, sc, ssum, v, aout, hout);
    zero_f<<<1, 256, 0, stream>>>(gsum, 3 * G_GRAPHS);
    relu_stats<<<nBlk, 256, 0, stream>>>(hout, batch, gsum, gsq, gcnt);
    finalize_stats<<<1, 64, 0, stream>>>(gsum, gsq, gcnt, gmean, ginv);
    apply_ln<<<nfBlk, 256, 0, stream>>>(hout, batch, gmean, ginv, gam, bet);
  };

  conv(x,  Wq1, bq1, Wk1, bk1, Wv1, bv1, Ws1, bs1, g1, be1, h1, alpha1);
  conv(h1, Wq2, bq2, Wk2, bk2, Wv2, bv2, Ws2, bs2, g2, be2, h2, alpha2);

  zero_f<<<(G_GRAPHS * HIDF + 255) / 256, 256, 0, stream>>>(feat, G_GRAPHS * HIDF);
  pool_acc<<<nfBlk, 256, 0, stream>>>(h2, batch, feat);
  pool_div<<<(G_GRAPHS * HIDF + 255) / 256, 256, 0, stream>>>(feat, gcnt);
}